// DifferentiableAAC_80023830659731
// MI455X (gfx1250) — compile-verified
//
#include <hip/hip_runtime.h>
#include <cmath>

typedef float v2f __attribute__((ext_vector_type(2)));
typedef float v8f __attribute__((ext_vector_type(8)));

#define NB 4
#define NC 2
#define NT 393216
#define NF 384
#define NR (NB*NC*NF)     /* 3072 long rows  */
#define NRS (NR*8)        /* 24576 short rows */
#define WINN 2048
#define KLONG 2048
#define NLONG 1024
#define KSHORT 256
#define NSHORT 128
#define KCHUNK 256
#define LDSTR 260         /* padded LDS row stride: %4==0 (float4 stores), %64==4 (no bank conflicts) */

static constexpr double PI_D = 3.14159265358979323846;

// ---------------- table generation (float64 on device, matching numpy) ----------------

__global__ void k_tab_long(float* __restrict__ CL, float* __restrict__ CLT) {
    int idx = blockIdx.x * 256 + threadIdx.x;          // 2048*1024
    int n = idx >> 10, k = idx & 1023;
    double c = cos(PI_D / 1024.0 * ((double)n + 512.5) * ((double)k + 0.5));
    float cf = (float)c;
    CL[idx] = cf;
    CLT[(size_t)k * WINN + n] = cf * (4.0f / 2048.0f); // exact pow2 scale
}

__global__ void k_tab_short(float* __restrict__ CS, float* __restrict__ CST) {
    int idx = blockIdx.x * 256 + threadIdx.x;          // 256*128
    int n = idx >> 7, k = idx & 127;
    double c = cos(PI_D / 128.0 * ((double)n + 64.5) * ((double)k + 0.5));
    float cf = (float)c;
    CS[idx] = cf;
    CST[(size_t)k * 256 + n] = cf * (4.0f / 256.0f);
}

__global__ void k_tab_win(float* __restrict__ WT, float* __restrict__ WS) {
    int n = blockIdx.x * 256 + threadIdx.x;            // 2048
    double wl = sin(PI_D * ((double)n + 0.5) / 2048.0);
    WT[0 * WINN + n] = (float)wl;
    WT[2 * WINN + n] = (float)wl;
    float ls;
    if (n < 1024)      ls = (float)wl;
    else if (n < 1472) ls = 1.0f;
    else if (n < 1600) ls = (float)sin(PI_D * ((double)(n - 1472 + 128) + 0.5) / 256.0);
    else               ls = 0.0f;
    WT[1 * WINN + n] = ls;
    float lp;
    if (n < 448)       lp = 0.0f;
    else if (n < 576)  lp = (float)sin(PI_D * ((double)(n - 448) + 0.5) / 256.0);
    else if (n < 1024) lp = 1.0f;
    else               lp = (float)wl;
    WT[3 * WINN + n] = lp;
    if (n < 256) WS[n] = (float)sin(PI_D * ((double)n + 0.5) / 256.0);
}

// ---------------- transients + window-sequence scan ----------------

__global__ void k_zero(int* __restrict__ p) {
    int i = blockIdx.x * 256 + threadIdx.x;
    if (i < NB * NF) p[i] = 0;
}

__global__ void __launch_bounds__(128) k_trans(const float* __restrict__ audio, int* __restrict__ trans) {
    __shared__ float sd[128];
    __shared__ float se[16];
    int bidx = blockIdx.x;                 // (b*C+c)*F + f
    int f = bidx % NF, bc = bidx / NF, b = bc / NC;
    int t = threadIdx.x;
    const float* arow = audio + (size_t)bc * NT + (size_t)f * 1024;
    bool lastf = (f == NF - 1);
    float p = 0.f;
    int base = t * 16;
    #pragma unroll 4
    for (int i = 0; i < 16; ++i) {
        int n = base + i;
        float v = (lastf && n >= 1024) ? 0.f : arow[n];
        p += v * v;
    }
    sd[t] = p;
    __syncthreads();
    if (t < 16) {
        float e = 0.f;
        for (int j = 0; j < 8; ++j) e += sd[t * 8 + j];
        se[t] = e;
    }
    __syncthreads();
    if (t == 0) {
        float mx = 0.f, mean = 0.f;
        for (int i = 0; i < 16; ++i) { float e = se[i]; mean += e; mx = fmaxf(mx, e); }
        mean *= (1.0f / 16.0f);
        float ratio = mx / (mean + 1e-9f);
        if (ratio > 8.0f && mx > 0.001f) atomicOr(&trans[b * NF + f], 1);
    }
}

__global__ void k_winseq(const int* __restrict__ trans, int* __restrict__ wseq) {
    int b = threadIdx.x;
    if (b >= NB) return;
    int prev = 0;  // ONLY_LONG
    for (int f = 0; f < NF; ++f) {
        int tr = trans[b * NF + f];
        bool was_long = (prev == 0) || (prev == 3);
        int ws = was_long ? (tr ? 1 : 0) : (tr ? 2 : 3);
        wseq[b * NF + f] = ws;
        prev = ws;
    }
}

// ---------------- gfx1250 async global->LDS staging (ASYNCcnt path) ----------------
// IOFFSET is added to BOTH the global address and the per-lane LDS address (ISA 10.x),
// so one base covers a thread's contiguous float4 run.

static __device__ __forceinline__ void async_copy_64B(unsigned int lds0, const float* gsrc) {
    asm volatile(
        "global_load_async_to_lds_b128 %0, %1, off\n"
        "global_load_async_to_lds_b128 %0, %1, off offset:16\n"
        "global_load_async_to_lds_b128 %0, %1, off offset:32\n"
        "global_load_async_to_lds_b128 %0, %1, off offset:48"
        :: "v"(lds0), "v"(gsrc) : "memory");
}

static __device__ __forceinline__ void async_copy_32B(unsigned int lds0, const float* gsrc) {
    asm volatile(
        "global_load_async_to_lds_b128 %0, %1, off\n"
        "global_load_async_to_lds_b128 %0, %1, off offset:16"
        :: "v"(lds0), "v"(gsrc) : "memory");
}

static __device__ __forceinline__ void wait_async0() {
    asm volatile("s_wait_asynccnt 0x0" ::: "memory");
}

// ---------------- fp32 WMMA inner K-loop (V_WMMA_F32_16X16X4_F32) ----------------
// A(16x4): lane<16 holds M=lane, K = comp + 2*(lane>>4).  B(4x16): K = comp + 2*half, N = lane&15.
// C/D(16x16): component v -> row v + 8*half, col lane&15.

static __device__ __forceinline__
v8f wmma_kchunk(const float* __restrict__ AsBase, const float* __restrict__ Bcol,
                int bstride, int kc, v8f acc, int l16, int half)
{
    const float* ap = AsBase + l16 * LDSTR + 2 * half;
    const float* bp = Bcol + (size_t)(2 * half) * bstride;
    for (int kb = 0; kb < kc; kb += 4) {
        v2f a = *(const v2f*)(ap + kb);
        v2f bb;
        bb.x = bp[(size_t)kb * bstride];
        bb.y = bp[(size_t)(kb + 1) * bstride];
        acc = __builtin_amdgcn_wmma_f32_16x16x4_f32(false, a, false, bb, (short)0, acc, false, false);
    }
    return acc;
}

// ---------------- long MDCT: coef[r,k] = sum_n frame[r,n]*w[wsq,n]*CL[n,k] ----------------

__global__ void __launch_bounds__(256) k_mdct_long(
    const float* __restrict__ audio, const int* __restrict__ wseq,
    const float* __restrict__ CL, const float* __restrict__ WT,
    float* __restrict__ out)
{
    __shared__ float As[16][LDSTR];
    int t = threadIdx.x;
    int m0 = blockIdx.x * 16;
    int wid = t >> 5, lane = t & 31, l16 = lane & 15, half = lane >> 4;
    int n0 = blockIdx.y * 128 + wid * 16;

    int srow = t >> 4;
    int cg = (t & 15) * 16;
    int r = m0 + srow;
    int f = r % NF, bc = r / NF, b = bc / NC;
    int wsq = wseq[b * NF + f];
    const float* arow = audio + (size_t)bc * NT + (size_t)f * 1024;
    const float* wrow = WT + (size_t)wsq * WINN;
    bool lastf = (f == NF - 1);

    v8f acc = {};
    for (int kc0 = 0; kc0 < KLONG; kc0 += KCHUNK) {
        #pragma unroll
        for (int j = 0; j < 4; ++j) {
            int col = cg + j * 4;
            int n = kc0 + col;
            float4 s = make_float4(0.f, 0.f, 0.f, 0.f);
            if (!(lastf && n >= 1024)) s = *(const float4*)(arow + n);
            float4 w4 = *(const float4*)(wrow + n);
            float4 d;
            d.x = s.x * w4.x; d.y = s.y * w4.y; d.z = s.z * w4.z; d.w = s.w * w4.w;
            *(float4*)&As[srow][col] = d;
        }
        __syncthreads();
        const float* Bcol = CL + (size_t)kc0 * NLONG + (n0 + l16);
        acc = wmma_kchunk(&As[0][0], Bcol, NLONG, KCHUNK, acc, l16, half);
        __syncthreads();
    }
    #pragma unroll
    for (int v = 0; v < 8; ++v) {
        int rr = m0 + v + 8 * half;
        out[(size_t)rr * NLONG + n0 + l16] = acc[v];
    }
}

// ---------------- short MDCT: blocks[rr,n] = frame(r, 448+128s+n)*WS[n];  x CS(256x128) ----------------

__global__ void __launch_bounds__(256) k_mdct_short(
    const float* __restrict__ audio, const float* __restrict__ CS,
    const float* __restrict__ WS, float* __restrict__ out)
{
    __shared__ float As[16][LDSTR];
    int t = threadIdx.x;
    int m0 = blockIdx.x * 16;
    int wid = t >> 5, lane = t & 31, l16 = lane & 15, half = lane >> 4;
    int n0 = wid * 16;

    int srow = t >> 4;
    int cg = (t & 15) * 16;
    int rr = m0 + srow;
    int r = rr >> 3, s = rr & 7;
    int f = r % NF, bc = r / NF;
    const float* arow = audio + (size_t)bc * NT + (size_t)f * 1024 + 448 + s * 128;
    bool lastf = (f == NF - 1);

    #pragma unroll
    for (int j = 0; j < 4; ++j) {
        int col = cg + j * 4;
        int n = 448 + s * 128 + col;
        float4 sv = make_float4(0.f, 0.f, 0.f, 0.f);
        if (!(lastf && n >= 1024)) sv = *(const float4*)(arow + col);
        float4 w4 = *(const float4*)(WS + col);
        float4 d;
        d.x = sv.x * w4.x; d.y = sv.y * w4.y; d.z = sv.z * w4.z; d.w = sv.w * w4.w;
        *(float4*)&As[srow][col] = d;
    }
    __syncthreads();
    v8f acc = {};
    const float* Bcol = CS + (n0 + l16);
    acc = wmma_kchunk(&As[0][0], Bcol, NSHORT, KSHORT, acc, l16, half);
    #pragma unroll
    for (int v = 0; v < 8; ++v)
        out[(size_t)(m0 + v + 8 * half) * NSHORT + n0 + l16] = acc[v];
}

// ---------------- select short coefficients into coef buffer ----------------

__global__ void k_select(const float* __restrict__ shortc, const int* __restrict__ wseq,
                         float* __restrict__ coef) {
    int idx = blockIdx.x * 256 + threadIdx.x;   // NR*1024
    int r = idx >> 10;
    int f = r % NF, b = (r / NF) / NC;
    if (wseq[b * NF + f] == 2) coef[idx] = shortc[idx];
}

// ---------------- per-(b,f) gain bisection (30 iters, block-wide bit-count reduce) ----------------

__global__ void __launch_bounds__(256) k_gains(const float* __restrict__ coef, float* __restrict__ gains) {
    __shared__ float sd[256];
    int bf = blockIdx.x;
    int b = bf / NF, f = bf % NF;
    int t = threadIdx.x;
    float xa[8];
    #pragma unroll
    for (int u = 0; u < 8; ++u) {
        int idx2 = t * 8 + u;
        int c = idx2 >> 10, k = idx2 & 1023;
        xa[u] = fabsf(coef[((size_t)((b * NC + c) * NF) + f) * 1024 + k]);
    }
    const float target = (float)(128000.0 * 1024.0 / 48000.0 / 2.0);
    float lo = -40.f, hi = 200.f;
    for (int it = 0; it < 30; ++it) {
        float mid = 0.5f * (lo + hi);
        float sc = exp2f(-mid * 0.25f);
        float bits = 0.f;
        #pragma unroll
        for (int u = 0; u < 8; ++u) {
            float s = xa[u] * sc;
            float q = 0.f;
            if (s > 0.f) q = floorf(powf(s, 0.75f) + 0.4054f);
            bits += 2.0f * log2f(1.0f + q);
        }
        sd[t] = bits;
        __syncthreads();
        for (int st = 128; st > 0; st >>= 1) {
            if (t < st) sd[t] += sd[t + st];
            __syncthreads();
        }
        float total = sd[0];
        __syncthreads();
        if (total <= target) hi = mid; else lo = mid;
    }
    if (t == 0) gains[bf] = rintf(hi);   // jnp.round = half-to-even
}

// ---------------- quantize(hard) + dequantize, in place ----------------

__global__ void k_qd(float* __restrict__ coef, const float* __restrict__ gains) {
    int idx = blockIdx.x * 256 + threadIdx.x;   // NR*1024
    int r = idx >> 10;
    int f = r % NF, b = (r / NF) / NC;
    float g = gains[b * NF + f];
    float x = coef[idx];
    float s = fabsf(x) * exp2f(-g * 0.25f);
    float comp = (s > 0.f) ? powf(s, 0.75f) : 0.f;
    float aq = floorf(comp + 0.4054f);
    float out = 0.f;
    if (aq > 0.f) out = copysignf(powf(aq, 4.0f / 3.0f) * exp2f(g * 0.25f), x);
    coef[idx] = out;
}

// ---------------- inverse long MDCT: rewin[r,n] = (rec x CLT)[r,n] * w[wsq,n] ----------------
// A tiles staged by async DMA (global_load_async_to_lds_b128), double-buffered so the
// chunk i+1 DMA overlaps the chunk i WMMA loop.

__global__ void __launch_bounds__(256) k_imdct_long(
    const float* __restrict__ rec, const int* __restrict__ wseq,
    const float* __restrict__ CLT, const float* __restrict__ WT,
    float* __restrict__ rewin)
{
    __shared__ float As[2][16][LDSTR];
    int t = threadIdx.x;
    int m0 = blockIdx.x * 16;
    int wid = t >> 5, lane = t & 31, l16 = lane & 15, half = lane >> 4;
    int n0 = blockIdx.y * 128 + wid * 16;

    int srow = t >> 4;
    int cg = (t & 15) * 16;
    const float* arow = rec + (size_t)(m0 + srow) * NLONG;
    unsigned int lds_b0 = (unsigned int)(uintptr_t)&As[0][srow][cg];
    unsigned int lds_b1 = (unsigned int)(uintptr_t)&As[1][srow][cg];

    async_copy_64B(lds_b0, arow + cg);          // chunk 0 -> buffer 0
    wait_async0();
    __syncthreads();

    v8f acc = {};
    #pragma unroll
    for (int ch = 0; ch < NLONG / KCHUNK; ++ch) {
        int buf = ch & 1;
        if (ch + 1 < NLONG / KCHUNK)            // prefetch chunk ch+1 into other buffer
            async_copy_64B(buf ? lds_b0 : lds_b1, arow + (ch + 1) * KCHUNK + cg);
        const float* Bcol = CLT + (size_t)(ch * KCHUNK) * WINN + (n0 + l16);
        acc = wmma_kchunk(&As[buf][0][0], Bcol, WINN, KCHUNK, acc, l16, half);
        wait_async0();
        __syncthreads();
    }
    int n = n0 + l16;
    #pragma unroll
    for (int v = 0; v < 8; ++v) {
        int rr = m0 + v + 8 * half;
        int f = rr % NF, b = (rr / NF) / NC;
        int wsq = wseq[b * NF + f];
        rewin[(size_t)rr * WINN + n] = acc[v] * WT[(size_t)wsq * WINN + n];
    }
}

// ---------------- inverse short MDCT: blk[rr,n] = (rec_s x CST)[rr,n] * WS[n] ----------------

__global__ void __launch_bounds__(256) k_imdct_short(
    const float* __restrict__ rec, const float* __restrict__ CST,
    const float* __restrict__ WS, float* __restrict__ blk)
{
    __shared__ float As[16][LDSTR];
    int t = threadIdx.x;
    int m0 = blockIdx.x * 16;
    int wid = t >> 5, lane = t & 31, l16 = lane & 15, half = lane >> 4;
    int n0 = blockIdx.y * 128 + wid * 16;

    int srow = t >> 4;
    int cg = (t & 15) * 8;
    const float* arow = rec + (size_t)(m0 + srow) * 128;  // (R,1024) == (R*8,128) linear
    unsigned int lds0 = (unsigned int)(uintptr_t)&As[srow][cg];
    async_copy_32B(lds0, arow + cg);
    wait_async0();
    __syncthreads();

    v8f acc = {};
    const float* Bcol = CST + (n0 + l16);
    acc = wmma_kchunk(&As[0][0], Bcol, 256, 128, acc, l16, half);
    int n = n0 + l16;
    float wn = WS[n];
    #pragma unroll
    for (int v = 0; v < 8; ++v)
        blk[(size_t)(m0 + v + 8 * half) * 256 + n] = acc[v] * wn;
}

// ---------------- in-frame overlap-add of the 8 short blocks (short frames only) ----------------

__global__ void __launch_bounds__(256) k_short_ov(const float* __restrict__ blk,
                                                  const int* __restrict__ wseq,
                                                  float* __restrict__ rewin) {
    int r = blockIdx.x;
    int f = r % NF, b = (r / NF) / NC;
    if (wseq[b * NF + f] != 2) return;
    int t = threadIdx.x;
    for (int j = 0; j < 8; ++j) {
        int n = t + j * 256;
        float acc = 0.f;
        #pragma unroll
        for (int s = 0; s < 8; ++s) {
            int off = n - (448 + s * 128);
            if (off >= 0 && off < 256) acc += blk[((size_t)r * 8 + s) * 256 + off];
        }
        rewin[(size_t)r * WINN + n] = acc;
    }
}

// ---------------- cross-frame overlap-add ----------------

__global__ void k_oadd(const float* __restrict__ rewin, float* __restrict__ out) {
    int idx = blockIdx.x * 256 + threadIdx.x;   // NB*NC*NT
    int bc = idx / NT, tt = idx % NT;
    int f = tt >> 10, i = tt & 1023;
    size_t r = (size_t)bc * NF + f;
    float v = rewin[r * WINN + i];
    if (f > 0) v += rewin[(r - 1) * WINN + 1024 + i];
    out[idx] = v;
}

// ---------------- launcher ----------------

extern "C" void kernel_launch(void* const* d_in, const int* in_sizes, int n_in,
                              void* d_out, int out_size, void* d_ws, size_t ws_size,
                              hipStream_t stream)
{
    const float* audio = (const float*)d_in[0];
    float* out = (float*)d_out;

    char* w = (char*)d_ws;
    size_t o = 0;
    auto carve = [&](size_t bytes) -> void* {
        void* p = w + o;
        o += (bytes + 255) & ~(size_t)255;
        return p;
    };
    float* CL    = (float*)carve((size_t)2048 * 1024 * 4);
    float* CLT   = (float*)carve((size_t)1024 * 2048 * 4);
    float* CS    = (float*)carve((size_t)256 * 128 * 4);
    float* CST   = (float*)carve((size_t)128 * 256 * 4);
    float* WT    = (float*)carve((size_t)4 * 2048 * 4);
    float* WS    = (float*)carve((size_t)256 * 4);
    float* coef  = (float*)carve((size_t)NR * 1024 * 4);
    float* csh   = (float*)carve((size_t)NR * 1024 * 4);
    float* rewin = (float*)carve((size_t)NR * 2048 * 4);
    float* blk   = (float*)carve((size_t)NRS * 256 * 4);
    float* gains = (float*)carve((size_t)NB * NF * 4);
    int*   trans = (int*)carve((size_t)NB * NF * 4);
    int*   wseq  = (int*)carve((size_t)NB * NF * 4);

    k_tab_long <<<(2048 * 1024) / 256, 256, 0, stream>>>(CL, CLT);
    k_tab_short<<<(256 * 128) / 256, 256, 0, stream>>>(CS, CST);
    k_tab_win  <<<2048 / 256, 256, 0, stream>>>(WT, WS);

    k_zero  <<<(NB * NF + 255) / 256, 256, 0, stream>>>(trans);
    k_trans <<<NR, 128, 0, stream>>>(audio, trans);
    k_winseq<<<1, 32, 0, stream>>>(trans, wseq);

    k_mdct_long <<<dim3(NR / 16, NLONG / 128), 256, 0, stream>>>(audio, wseq, CL, WT, coef);
    k_mdct_short<<<NRS / 16, 256, 0, stream>>>(audio, CS, WS, csh);
    k_select    <<<(NR * 1024) / 256, 256, 0, stream>>>(csh, wseq, coef);

    k_gains<<<NB * NF, 256, 0, stream>>>(coef, gains);
    k_qd   <<<(NR * 1024) / 256, 256, 0, stream>>>(coef, gains);

    k_imdct_long <<<dim3(NR / 16, WINN / 128), 256, 0, stream>>>(coef, wseq, CLT, WT, rewin);
    k_imdct_short<<<dim3(NRS / 16, 2), 256, 0, stream>>>(coef, CST, WS, blk);
    k_short_ov   <<<NR, 256, 0, stream>>>(blk, wseq, rewin);
    k_oadd       <<<(NB * NC * NT) / 256, 256, 0, stream>>>(rewin, out);
}